// MultiHeadAttention_5463198401093
// MI455X (gfx1250) — compile-verified
//
#include <hip/hip_runtime.h>
#include <math.h>

typedef __attribute__((ext_vector_type(2))) float v2f;
typedef __attribute__((ext_vector_type(8))) float v8f;
typedef int i32x4 __attribute__((vector_size(16)));

#define B_  4
#define S_  2048
#define D_  1024
#define H_  16
#define DK_ 64

static constexpr float NEGV = -1000000000.0f;
static constexpr float EPSV = 1e-5f;

// ---- CDNA5 async global->LDS copy (ASYNCcnt path), with portable fallback --
#if __has_builtin(__builtin_amdgcn_global_load_async_to_lds_b128)
#define HAVE_ASYNC_LDS 1
#else
#define HAVE_ASYNC_LDS 0
#endif

__device__ __forceinline__ void async_copy16(const float* g, float* l) {
#if HAVE_ASYNC_LDS
    __builtin_amdgcn_global_load_async_to_lds_b128(
        (__attribute__((address_space(1))) i32x4*)(g),
        (__attribute__((address_space(3))) i32x4*)(l), 0, 0);
#else
    *(float4*)l = *(const float4*)g;
#endif
}

__device__ __forceinline__ void async_wait() {
#if HAVE_ASYNC_LDS
#if __has_builtin(__builtin_amdgcn_s_wait_asynccnt)
    __builtin_amdgcn_s_wait_asynccnt(0);
#else
    asm volatile("s_wait_asynccnt 0x0" ::: "memory");
#endif
#endif
}

// D = A(16x4) * B(4x16) + C, full f32. Wave32, EXEC all ones where called.
__device__ __forceinline__ v8f wmma_f32(v2f a, v2f b, v8f c) {
    return __builtin_amdgcn_wmma_f32_16x16x4_f32(
        /*neg_a=*/false, a, /*neg_b=*/false, b,
        /*c_mod=*/(short)0, c, /*reuse_a=*/false, /*reuse_b=*/false);
}

// LDS row strides (floats): multiple of 4 (16B-aligned granules), not a
// multiple of 64 banks (36 % 64 spreads rows across banks).
#define KPAD 36
#define SROW 2052
#define QROW 68

// -------------------------------------------------------------------------
// Kernel 1: fused QKV projection.  out = X @ W^T + b, written head-split.
// grid: (B*S/64, D/32, 3)   block: 256 (8 waves; 4x2 grid of 16x16 tiles)
// -------------------------------------------------------------------------
__global__ __launch_bounds__(256)
void qkv_gemm_kernel(const float* __restrict__ q_in,
                     const float* __restrict__ k_in,
                     const float* __restrict__ v_in,
                     const float* __restrict__ Wq, const float* __restrict__ bq,
                     const float* __restrict__ Wk, const float* __restrict__ bk,
                     const float* __restrict__ Wv, const float* __restrict__ bv,
                     float* __restrict__ Qh, float* __restrict__ Kh,
                     float* __restrict__ Vh)
{
    const int which = blockIdx.z;
    const float* X    = (which == 0) ? q_in : (which == 1) ? k_in : v_in;
    const float* W    = (which == 0) ? Wq   : (which == 1) ? Wk   : Wv;
    const float* bias = (which == 0) ? bq   : (which == 1) ? bk   : bv;
    float*       dst  = (which == 0) ? Qh   : (which == 1) ? Kh   : Vh;

    __shared__ float As[64][KPAD];   // [m][k] tile of X
    __shared__ float Bs[32][KPAD];   // [n][k] tile of W (row n contiguous in k)

    const int tid  = threadIdx.x;
    const int lane = tid & 31;
    const int wid  = tid >> 5;      // 0..7
    const int wm   = wid & 3;       // M subtile
    const int wn   = wid >> 2;      // N subtile (0..1)
    const int half = lane >> 4;     // 0/1
    const int r    = lane & 15;

    const int m0 = blockIdx.x * 64;
    const int n0 = blockIdx.y * 32;

    v8f acc = {};

    for (int k0 = 0; k0 < D_; k0 += 32) {
        // async staging: As = 64x32 (512 granules, 2/thread), Bs = 32x32 (1/thread)
        #pragma unroll
        for (int i = 0; i < 2; ++i) {
            const int e   = (tid + i * 256) * 4;
            const int row = e >> 5;
            const int col = e & 31;
            async_copy16(&X[(size_t)(m0 + row) * D_ + k0 + col], &As[row][col]);
        }
        {
            const int e   = tid * 4;
            const int row = e >> 5;
            const int col = e & 31;
            async_copy16(&W[(size_t)(n0 + row) * D_ + k0 + col], &Bs[row][col]);
        }
        async_wait();
        __syncthreads();

        #pragma unroll
        for (int kk = 0; kk < 32; kk += 4) {
            v2f a, b;
            a.x = As[wm * 16 + r][kk + 2 * half];
            a.y = As[wm * 16 + r][kk + 2 * half + 1];
            // B[k][c] = W[n0 + c][k] = Bs[c][k]
            b.x = Bs[wn * 16 + r][kk + 2 * half];
            b.y = Bs[wn * 16 + r][kk + 2 * half + 1];
            acc = wmma_f32(a, b, acc);
        }
        __syncthreads();
    }

    // epilogue: bias + head-split store  [B,H,S,DK]
    const int n_g = n0 + wn * 16 + r;
    const int h   = n_g / DK_;
    const int dk  = n_g % DK_;
    const float bval = bias[n_g];
    #pragma unroll
    for (int i = 0; i < 8; ++i) {
        const int m_g = m0 + wm * 16 + i + 8 * half;
        const int bb  = m_g / S_;
        const int ss  = m_g % S_;
        dst[(((size_t)bb * H_ + h) * S_ + ss) * DK_ + dk] = acc[i] + bval;
    }
}

// -------------------------------------------------------------------------
// Kernel 2: attention.  One block = 16 query rows of one (b,h).
// grid: (S/16, B*H)   block: 256 (8 waves)
// LDS: scores[16][2052] (full 16xS row block) + qs[16][68] + reductions.
// -------------------------------------------------------------------------
__global__ __launch_bounds__(256)
void attn_kernel(const float* __restrict__ Qh, const float* __restrict__ Kh,
                 const float* __restrict__ Vh, const int* __restrict__ mask,
                 float* __restrict__ attn_out, float* __restrict__ O)
{
    extern __shared__ float sm[];
    float* scores = sm;                      // 16 * SROW
    float* qs     = scores + 16 * SROW;      // 16 * QROW (reused as split-K scratch)
    float* redm   = qs + 16 * QROW;          // 16 * 16
    float* reds   = redm + 16 * 16;          // 16 * 16

    const int tid  = threadIdx.x;
    const int lane = tid & 31;
    const int wid  = tid >> 5;   // 0..7
    const int half = lane >> 4;
    const int r    = lane & 15;

    const int qt   = blockIdx.x;
    const int bh   = blockIdx.y;        // b*H + h
    const int bidx = bh / H_;
    const int h    = bh % H_;
    const int q0   = qt * 16;

    const float* Qp = Qh + (size_t)bh * S_ * DK_;
    const float* Kp = Kh + (size_t)bh * S_ * DK_;
    const float* Vp = Vh + (size_t)bh * S_ * DK_;

    // stage Q tile (16 x 64) via async b128 (256 granules, 1/thread)
    {
        const int e  = tid * 4;
        const int rr = e >> 6;
        const int cc = e & 63;
        async_copy16(&Qp[(size_t)(q0 + rr) * DK_ + cc], &qs[rr * QROW + cc]);
    }
    async_wait();
    __syncthreads();

    // scores = Q K^T / sqrt(DK); 8 waves stride over 128 key tiles
    for (int jt = wid; jt < S_ / 16; jt += 8) {
        const int j0 = jt * 16;
        v8f acc = {};
        #pragma unroll
        for (int d = 0; d < DK_; d += 4) {
            v2f a, b;
            a.x = qs[r * QROW + d + 2 * half];
            a.y = qs[r * QROW + d + 2 * half + 1];
            // B[k][c] = K[j0+c][d+k]; two adjacent floats -> b64 load
            const float2 kb = *(const float2*)&Kp[(size_t)(j0 + r) * DK_ + d + 2 * half];
            b.x = kb.x;
            b.y = kb.y;
            acc = wmma_f32(a, b, acc);
        }
        #pragma unroll
        for (int i = 0; i < 8; ++i) {
            const int row = i + 8 * half;
            scores[row * SROW + j0 + r] = acc[i] * 0.125f;  // 1/sqrt(64)
        }
    }
    __syncthreads();

    // masked softmax over the full 16 x S block in LDS (16 threads per row)
    const int row = tid >> 4;          // 0..15
    const int sub = tid & 15;          // 16 threads per row
    const int qg  = q0 + row;
    const int* mrow = mask + ((size_t)bidx * S_ + qg) * S_;
    float* srow = scores + row * SROW;

    float mx = NEGV;
    for (int c = sub; c < S_; c += 16) {
        const float v = (mrow[c] == 0) ? NEGV : srow[c];
        srow[c] = v;
        mx = fmaxf(mx, v);
    }
    redm[row * 16 + sub] = mx;
    __syncthreads();
    if (sub == 0) {
        float m2 = redm[row * 16];
        #pragma unroll
        for (int i = 1; i < 16; ++i) m2 = fmaxf(m2, redm[row * 16 + i]);
        redm[row * 16] = m2;
    }
    __syncthreads();
    mx = redm[row * 16];

    float sum = 0.f;
    for (int c = sub; c < S_; c += 16) {
        const float e = __expf(srow[c] - mx);
        srow[c] = e;
        sum += e;
    }
    reds[row * 16 + sub] = sum;
    __syncthreads();
    if (sub == 0) {
        float s2 = 0.f;
        #pragma unroll
        for (int i = 0; i < 16; ++i) s2 += reds[row * 16 + i];
        reds[row * 16] = 1.0f / s2;
    }
    __syncthreads();
    const float inv = reds[row * 16];

    float* arow = attn_out + ((size_t)bh * S_ + qg) * S_;
    for (int c = sub; c < S_; c += 16) {
        const float p = srow[c] * inv;
        srow[c] = p;          // reuse as A-matrix for attn @ V
        arow[c] = p;          // stream 1 GiB attn tensor out
    }
    __syncthreads();

    // out = attn @ V with split-K over 8 waves:
    // wave w handles dk columns [(w&3)*16, +16) over K half (w>>2).
    {
        const int n0v   = (wid & 3) * 16;
        const int kbase = (wid >> 2) * (S_ / 2);
        v8f acc = {};
        for (int kk = kbase; kk < kbase + S_ / 2; kk += 4) {
            v2f a, b;
            a.x = scores[r * SROW + kk + 2 * half];
            a.y = scores[r * SROW + kk + 2 * half + 1];
            b.x = Vp[(size_t)(kk + 2 * half) * DK_ + n0v + r];
            b.y = Vp[(size_t)(kk + 2 * half + 1) * DK_ + n0v + r];
            acc = wmma_f32(a, b, acc);
        }
        float* part = qs;   // Q tile dead; 4*256 floats <= 16*QROW
        if (wid >= 4) {
            #pragma unroll
            for (int i = 0; i < 8; ++i)
                part[(wid & 3) * 256 + (i + 8 * half) * 16 + r] = acc[i];
        }
        __syncthreads();
        if (wid < 4) {
            #pragma unroll
            for (int i = 0; i < 8; ++i) {
                const float tot = acc[i] + part[wid * 256 + (i + 8 * half) * 16 + r];
                const int qg2 = q0 + i + 8 * half;
                O[((size_t)bidx * S_ + qg2) * D_ + h * DK_ + n0v + r] = tot;
            }
        }
    }
}

// -------------------------------------------------------------------------
// Kernel 3: out = O @ Wo^T + bo + residual
// grid: (B*S/64, D/32)   block: 256
// -------------------------------------------------------------------------
__global__ __launch_bounds__(256)
void out_proj_kernel(const float* __restrict__ O, const float* __restrict__ Wo,
                     const float* __restrict__ bo, const float* __restrict__ resid,
                     float* __restrict__ X)
{
    __shared__ float As[64][KPAD];
    __shared__ float Bs[32][KPAD];

    const int tid  = threadIdx.x;
    const int lane = tid & 31;
    const int wid  = tid >> 5;
    const int wm   = wid & 3;
    const int wn   = wid >> 2;
    const int half = lane >> 4;
    const int r    = lane & 15;

    const int m0 = blockIdx.x * 64;
    const int n0 = blockIdx.y * 32;

    v8f acc = {};

    for (int k0 = 0; k0 < D_; k0 += 32) {
        #pragma unroll
        for (int i = 0; i < 2; ++i) {
            const int e   = (tid + i * 256) * 4;
            const int row = e >> 5;
            const int col = e & 31;
            async_copy16(&O[(size_t)(m0 + row) * D_ + k0 + col], &As[row][col]);
        }
        {
            const int e   = tid * 4;
            const int row = e >> 5;
            const int col = e & 31;
            async_copy16(&Wo[(size_t)(n0 + row) * D_ + k0 + col], &Bs[row][col]);
        }
        async_wait();
        __syncthreads();

        #pragma unroll
        for (int kk = 0; kk < 32; kk += 4) {
            v2f a, b;
            a.x = As[wm * 16 + r][kk + 2 * half];
            a.y = As[wm * 16 + r][kk + 2 * half + 1];
            b.x = Bs[wn * 16 + r][kk + 2 * half];
            b.y = Bs[wn * 16 + r][kk + 2 * half + 1];
            acc = wmma_f32(a, b, acc);
        }
        __syncthreads();
    }

    const int n_g = n0 + wn * 16 + r;
    const float bval = bo[n_g];
    #pragma unroll
    for (int i = 0; i < 8; ++i) {
        const int m_g = m0 + wm * 16 + i + 8 * half;
        const size_t idx = (size_t)m_g * D_ + n_g;
        X[idx] = acc[i] + bval + resid[idx];
    }
}

// -------------------------------------------------------------------------
// Kernel 4: row LayerNorm (D = 1024).  grid: B*S, block: 256
// -------------------------------------------------------------------------
__global__ __launch_bounds__(256)
void layernorm_kernel(const float* __restrict__ X, const float* __restrict__ gamma,
                      const float* __restrict__ beta, float* __restrict__ Y)
{
    __shared__ float red[256];
    const int tid = threadIdx.x;
    const float* xr = X + (size_t)blockIdx.x * D_;

    float vals[4];
    float s = 0.f;
    #pragma unroll
    for (int j = 0; j < 4; ++j) { vals[j] = xr[tid + j * 256]; s += vals[j]; }
    red[tid] = s;
    __syncthreads();
    for (int off = 128; off > 0; off >>= 1) {
        if (tid < off) red[tid] += red[tid + off];
        __syncthreads();
    }
    const float mu = red[0] * (1.0f / D_);
    __syncthreads();

    float v = 0.f;
    #pragma unroll
    for (int j = 0; j < 4; ++j) { const float d = vals[j] - mu; v += d * d; }
    red[tid] = v;
    __syncthreads();
    for (int off = 128; off > 0; off >>= 1) {
        if (tid < off) red[tid] += red[tid + off];
        __syncthreads();
    }
    const float rstd = rsqrtf(red[0] * (1.0f / D_) + EPSV);

    float* yr = Y + (size_t)blockIdx.x * D_;
    #pragma unroll
    for (int j = 0; j < 4; ++j) {
        const int c = tid + j * 256;
        yr[c] = (vals[j] - mu) * rstd * gamma[c] + beta[c];
    }
}

// -------------------------------------------------------------------------
extern "C" void kernel_launch(void* const* d_in, const int* in_sizes, int n_in,
                              void* d_out, int out_size, void* d_ws, size_t ws_size,
                              hipStream_t stream) {
    (void)in_sizes; (void)n_in; (void)out_size; (void)ws_size;

    const float* query = (const float*)d_in[0];
    const float* key   = (const float*)d_in[1];
    const float* value = (const float*)d_in[2];
    const int*   mask  = (const int*)d_in[3];
    const float* Wq = (const float*)d_in[4];  const float* bq = (const float*)d_in[5];
    const float* Wk = (const float*)d_in[6];  const float* bk = (const float*)d_in[7];
    const float* Wv = (const float*)d_in[8];  const float* bv = (const float*)d_in[9];
    const float* Wo = (const float*)d_in[10]; const float* bo = (const float*)d_in[11];
    const float* gamma = (const float*)d_in[12];
    const float* beta  = (const float*)d_in[13];

    float* y_out    = (float*)d_out;                                  // (B,S,D)
    float* attn_out = y_out + (size_t)B_ * S_ * D_;                   // (B,H,S,S)

    const size_t BHSDK = (size_t)B_ * H_ * S_ * DK_;  // 8,388,608 floats
    float* ws = (float*)d_ws;
    float* Qh = ws;
    float* Kh = ws + BHSDK;
    float* Vh = ws + 2 * BHSDK;
    float* O  = ws + 3 * BHSDK;
    float* X  = ws;            // reuse Qh region after attention

    // 1) QKV projections
    qkv_gemm_kernel<<<dim3(B_ * S_ / 64, D_ / 32, 3), dim3(256), 0, stream>>>(
        query, key, value, Wq, bq, Wk, bk, Wv, bv, Qh, Kh, Vh);

    // 2) attention (single-pass softmax: full 16xS score block in 320KB LDS)
    const size_t lds_bytes =
        (size_t)(16 * SROW + 16 * QROW + 16 * 16 * 2) * sizeof(float);
    attn_kernel<<<dim3(S_ / 16, B_ * H_), dim3(256), lds_bytes, stream>>>(
        Qh, Kh, Vh, mask, attn_out, O);

    // 3) output projection + residual
    out_proj_kernel<<<dim3(B_ * S_ / 64, D_ / 32), dim3(256), 0, stream>>>(
        O, Wo, bo, query, X);

    // 4) layernorm
    layernorm_kernel<<<dim3(B_ * S_), dim3(256), 0, stream>>>(X, gamma, beta, y_out);
}